// MoLELayer_68573447848335
// MI455X (gfx1250) — compile-verified
//
#include <hip/hip_runtime.h>
#include <math.h>

#define B_ 8
#define L_ 4096
#define D_ 2048
#define E_ 8
#define TOPK_ 2
#define RANK_ 8
#define LSPLIT 16
#define LCHUNK (L_ / LSPLIT)          // 256 rows per partial-sum chunk
#define HPAD 16                       // h padded to 16 rows for branch-free WMMA B-frags
#define LN_EPS 1e-5f
#define ALPHA_ (1.0f / (float)RANK_)

typedef float v2f __attribute__((ext_vector_type(2)));
typedef float v8f __attribute__((ext_vector_type(8)));

// ---------------------------------------------------------------------------
// K1: partial column sums over L (deterministic two-phase reduction, no atomics)
// grid = (D/256, B, LSPLIT), block = 256
// ---------------------------------------------------------------------------
__global__ void k_poolpart(const float* __restrict__ x, float* __restrict__ part) {
    int d = blockIdx.x * 256 + threadIdx.x;   // 0..D-1
    int b = blockIdx.y;                       // 0..B-1
    int c = blockIdx.z;                       // 0..LSPLIT-1
    const float* xp = x + ((size_t)b * L_ + (size_t)c * LCHUNK) * D_ + d;
    float s = 0.f;
    #pragma unroll 8
    for (int l = 0; l < LCHUNK; ++l) s += xp[(size_t)l * D_];
    part[((size_t)c * B_ + b) * D_ + d] = s;
}

// ---------------------------------------------------------------------------
// K2: finalize h (padded to 16 rows, rows 8..15 = 0), router logits,
//     top-2 + softmax weights, aux losses. Single block of 256 threads.
// ---------------------------------------------------------------------------
__global__ void k_router(const float* __restrict__ part,
                         const float* __restrict__ gate_w,
                         const float* __restrict__ gate_b,
                         float* __restrict__ h,        // [HPAD, D]
                         float* __restrict__ wts,      // [B, 2]
                         int*   __restrict__ eidx,     // [B, 2]
                         float* __restrict__ aux_out)  // scalar slot
{
    __shared__ float lg[B_ * E_];
    __shared__ float sm[B_ * E_];
    __shared__ float lse2[B_];
    __shared__ int   se[B_ * TOPK_];
    int tid = threadIdx.x;

    // finalize h = sum(partials, fixed order) / L ; zero the pad rows
    for (int i = tid; i < B_ * D_; i += 256) {
        float s = 0.f;
        #pragma unroll
        for (int c = 0; c < LSPLIT; ++c) s += part[(size_t)c * (B_ * D_) + i];
        h[i] = s / (float)L_;
    }
    for (int i = B_ * D_ + tid; i < HPAD * D_; i += 256) h[i] = 0.f;
    __syncthreads();

    // logits[b][e] = h[b] . gate_w[e] + gate_b[e]
    if (tid < B_ * E_) {
        int b = tid >> 3, e = tid & 7;
        const float* hp = h + (size_t)b * D_;
        const float* wp = gate_w + (size_t)e * D_;
        float acc = 0.f;
        for (int d = 0; d < D_; ++d) acc += hp[d] * wp[d];
        lg[b * E_ + e] = acc + gate_b[e];
    }
    __syncthreads();

    // per-sequence top-2, softmax weights, full softmax (for P), logsumexp^2
    if (tid < B_) {
        int b = tid;
        float best = -3.0e38f; int bi = 0;
        for (int e = 0; e < E_; ++e) {
            float v = lg[b * E_ + e];
            if (v > best) { best = v; bi = e; }
        }
        float sec = -3.0e38f; int si = 0;
        for (int e = 0; e < E_; ++e) {
            if (e == bi) continue;
            float v = lg[b * E_ + e];
            if (v > sec) { sec = v; si = e; }
        }
        float e1 = expf(sec - best);
        float inv = 1.0f / (1.0f + e1);
        se[b * 2 + 0] = bi;  se[b * 2 + 1] = si;
        eidx[b * 2 + 0] = bi; eidx[b * 2 + 1] = si;
        wts[b * 2 + 0] = inv;
        wts[b * 2 + 1] = e1 * inv;

        float den = 0.f;
        for (int e = 0; e < E_; ++e) den += expf(lg[b * E_ + e] - best);
        for (int e = 0; e < E_; ++e) sm[b * E_ + e] = expf(lg[b * E_ + e] - best) / den;
        float lse = best + logf(den);
        lse2[b] = lse * lse;
    }
    __syncthreads();

    // aux = 0.01*load_balance + 0.001*z_loss + 0.01*renyi
    if (tid == 0) {
        float counts[E_];
        for (int e = 0; e < E_; ++e) counts[e] = 0.f;
        for (int i = 0; i < B_ * TOPK_; ++i) counts[se[i]] += 1.f;
        float lb = 0.f, f[E_];
        for (int e = 0; e < E_; ++e) {
            f[e] = counts[e] / (float)(B_ * TOPK_);
            float pe = 0.f;
            for (int b = 0; b < B_; ++b) pe += sm[b * E_ + e];
            lb += f[e] * (pe / (float)B_);
        }
        lb *= (float)E_;
        float z = 0.f;
        for (int b = 0; b < B_; ++b) z += lse2[b];
        z /= (float)B_;
        float psum = 0.f;
        for (int e = 0; e < E_; ++e) psum += f[e] + 1e-8f;
        float s2 = 0.f;
        for (int e = 0; e < E_; ++e) { float p = (f[e] + 1e-8f) / psum; s2 += p * p; }
        aux_out[0] = 0.01f * lb + 0.001f * z + 0.01f * logf(s2);
    }
}

// ---------------------------------------------------------------------------
// K3: T[er, b] = sum_d A[er, d] * h[b, d]  via V_WMMA_F32_16X16X4_F32
// A combined = [E*RANK, D] = [64, 2048]; h padded to [16, D] so every lane
// loads unconditionally (no EXEC divergence in the hot loop).
// One block, 4 waves; wave w owns rows [16w, 16w+16), K-loop in steps of 4.
// ---------------------------------------------------------------------------
__global__ void k_lora_t(const float* __restrict__ Amat,  // [64, D]
                         const float* __restrict__ h,     // [HPAD, D]
                         float* __restrict__ T)           // [64, 16]
{
    int lane  = threadIdx.x & 31;
    int wave  = threadIdx.x >> 5;     // 0..3
    int lmod  = lane & 15;
    int lhalf = lane >> 4;            // 0: K={0,1}; 1: K={2,3}  (ISA 16x4 f32 A layout)

    const float* ap = Amat + (size_t)(wave * 16 + lmod) * D_ + 2 * lhalf;
    const float* hp = h    + (size_t)lmod * D_ + 2 * lhalf;   // pad rows are zero

    v8f acc = {};
    for (int k0 = 0; k0 < D_; k0 += 4) {
        v2f a  = *(const v2f*)(ap + k0);
        v2f bm = *(const v2f*)(hp + k0);
        acc = __builtin_amdgcn_wmma_f32_16x16x4_f32(
            /*neg_a=*/false, a, /*neg_b=*/false, bm,
            /*c_mod=*/(short)0, acc, /*reuse_a=*/false, /*reuse_b=*/false);
    }
    // C/D layout: VGPR j -> rows M=j (lanes 0-15) and M=j+8 (lanes 16-31), N=lane%16
    #pragma unroll
    for (int j = 0; j < 8; ++j) {
        int m = wave * 16 + j + 8 * lhalf;
        T[m * 16 + lmod] = acc[j];
    }
}

// ---------------------------------------------------------------------------
// K4: delta[b,d] = ALPHA * sum_k w[b,k] * sum_r Bw[e_k, d, r] * T[e_k*8 + r, b]
// grid = (D/256, B), block = 256
// ---------------------------------------------------------------------------
__global__ void k_delta(const float* __restrict__ Bw,    // [E, D, RANK]
                        const float* __restrict__ T,     // [64, 16]
                        const float* __restrict__ wts,   // [B, 2]
                        const int*   __restrict__ eidx,  // [B, 2]
                        float* __restrict__ delta)       // [B, D]
{
    int d = blockIdx.x * 256 + threadIdx.x;
    int b = blockIdx.y;
    float acc = 0.f;
    #pragma unroll
    for (int k = 0; k < TOPK_; ++k) {
        int   e = eidx[b * 2 + k];
        float w = wts[b * 2 + k];
        const float* bp = Bw + ((size_t)e * D_ + d) * RANK_;   // 8 contiguous floats
        float s = 0.f;
        #pragma unroll
        for (int r = 0; r < RANK_; ++r) s += bp[r] * T[(e * RANK_ + r) * 16 + b];
        acc += w * s;
    }
    delta[(size_t)b * D_ + d] = acc * ALPHA_;
}

// ---------------------------------------------------------------------------
// K5: y = LayerNorm(x + delta) * gamma + beta  — the bandwidth-bound kernel.
// One row (b,l) per block; stage v = x+delta in LDS so x is read once.
// grid = B*L, block = 256
// ---------------------------------------------------------------------------
__global__ void k_ln(const float* __restrict__ x,
                     const float* __restrict__ delta,
                     const float* __restrict__ gamma,
                     const float* __restrict__ beta,
                     float* __restrict__ y)
{
    __shared__ float vbuf[D_];
    __shared__ float rs[256];
    __shared__ float rq[256];
    int row = blockIdx.x;
    int b   = row >> 12;                       // row / L_ (L_ = 4096)
    const float* xr = x + (size_t)row * D_;
    const float* dr = delta + (size_t)b * D_;
    int tid = threadIdx.x;

    float s = 0.f, q = 0.f;
    #pragma unroll
    for (int i = tid; i < D_; i += 256) {
        float v = xr[i] + dr[i];
        vbuf[i] = v;
        s += v;
        q += v * v;
    }
    rs[tid] = s; rq[tid] = q;
    __syncthreads();
    #pragma unroll
    for (int off = 128; off > 0; off >>= 1) {
        if (tid < off) { rs[tid] += rs[tid + off]; rq[tid] += rq[tid + off]; }
        __syncthreads();
    }
    float mu   = rs[0] * (1.0f / (float)D_);
    float var  = rq[0] * (1.0f / (float)D_) - mu * mu;
    float rstd = rsqrtf(var + LN_EPS);

    float* yr = y + (size_t)row * D_;
    #pragma unroll
    for (int i = tid; i < D_; i += 256) {
        yr[i] = (vbuf[i] - mu) * rstd * gamma[i] + beta[i];
    }
}

// ---------------------------------------------------------------------------
// launch
// ---------------------------------------------------------------------------
extern "C" void kernel_launch(void* const* d_in, const int* in_sizes, int n_in,
                              void* d_out, int out_size, void* d_ws, size_t ws_size,
                              hipStream_t stream) {
    (void)in_sizes; (void)n_in; (void)out_size; (void)ws_size;
    const float* x      = (const float*)d_in[0];
    const float* gate_w = (const float*)d_in[1];
    const float* gate_b = (const float*)d_in[2];
    const float* Amat   = (const float*)d_in[3];   // [E, RANK, D] == [64, D]
    const float* Bw     = (const float*)d_in[4];   // [E, D, RANK]
    const float* gamma  = (const float*)d_in[5];
    const float* beta   = (const float*)d_in[6];

    float* y   = (float*)d_out;                    // [B, L, D]
    float* aux = y + (size_t)B_ * L_ * D_;         // scalar tail

    float* ws    = (float*)d_ws;
    float* part  = ws;                             // LSPLIT*B*D = 262144 f
    float* h     = part + (size_t)LSPLIT * B_ * D_;// HPAD*D = 32768 f (rows 8..15 zero)
    float* T     = h + (size_t)HPAD * D_;          // 64*16 = 1024 f
    float* wts   = T + 64 * 16;                    // B*2 = 16 f
    float* delta = wts + B_ * 2;                   // B*D = 16384 f
    int*   eidx  = (int*)(delta + B_ * D_);        // B*2 ints

    dim3 g1(D_ / 256, B_, LSPLIT);
    k_poolpart<<<g1, 256, 0, stream>>>(x, part);

    k_router<<<1, 256, 0, stream>>>(part, gate_w, gate_b, h, wts, eidx, aux);

    k_lora_t<<<1, 128, 0, stream>>>(Amat, h, T);

    dim3 g4(D_ / 256, B_);
    k_delta<<<g4, 256, 0, stream>>>(Bw, T, wts, eidx, delta);

    k_ln<<<B_ * L_, 256, 0, stream>>>(x, delta, gamma, beta, y);
}